// ProposalFilter_34815004901958
// MI455X (gfx1250) — compile-verified
//
#include <hip/hip_runtime.h>
#include <hip/hip_bf16.h>
#include <stdint.h>

#define NPROP   4096
#define NBATCH  16
#define MAXP    300
#define NTHREAD 1024
#define IOU_THRESH   0.7f
#define SCORE_THRESH 0.5f

typedef unsigned int u32x4 __attribute__((ext_vector_type(4)));
typedef int          i32x8 __attribute__((ext_vector_type(8)));
typedef int          i32x4 __attribute__((ext_vector_type(4)));

__global__ __launch_bounds__(NTHREAD)
void nms_gfx1250_kernel(const float* __restrict__ proposals,   // [B, N, 4]
                        const float* __restrict__ cls_scores,  // [B, N, 2]
                        float* __restrict__ out)               // boxes then scores
{
    // Whole per-image working set stays resident in the WGP's 320KB LDS.
    __shared__ float    s_scr[NPROP * 2];             // 32 KB  raw cls_scores[b] (TDM-staged)
    __shared__ float    s_key[NPROP];                 // 16 KB  sort keys
    __shared__ int      s_idx[NPROP];                 // 16 KB  original indices ("order")
    __shared__ __align__(16) float s_box[NPROP][4];   // 64 KB  sorted boxes (AoS)
    __shared__ float    s_area[NPROP];                // 16 KB
    __shared__ unsigned s_supp[NPROP / 32];           // 512 B  suppression bitmap
    __shared__ int      s_sel[MAXP];                  // selected sorted-positions

    const int b   = blockIdx.x;
    const int tid = threadIdx.x;
    const float* props = proposals  + (size_t)b * NPROP * 4;
    const float* scr   = cls_scores + (size_t)b * NPROP * 2;

    // gfx1250: global_prefetch_b8 to warm L2 for the box gather
    __builtin_prefetch(props, 0, 0);

    // ---- 0. TDM: stage the whole 4096x2 f32 score block into LDS -----------
    // One per-wave DMA (wave 0 only; other waves branch around it since TDM
    // ignores EXEC). D# per CDNA5 ISA ch.8: 1-D tensor, 8192 elems of 4B.
    if (tid < 32) {
        uint64_t ga  = (uint64_t)(uintptr_t)scr;
        unsigned lds = (unsigned)(uintptr_t)(&s_scr[0]);   // wave-relative LDS byte addr

        u32x4 g0;
        g0[0] = 1u;                                        // count=1, is_restore=0
        g0[1] = lds;                                       // lds_addr
        g0[2] = (unsigned)(ga & 0xFFFFFFFFu);              // global_addr[31:0]
        g0[3] = (unsigned)((ga >> 32) & 0x01FFFFFFu)       // global_addr[56:32]
              | (2u << 30);                                // type=2 ("image")

        const unsigned TD0 = NPROP * 2;                    // tensor_dim0 = 8192 elems
        i32x8 g1;
        g1[0] = (int)(2u << 16);                           // wg_mask=0, data_size=4B
        g1[1] = (int)((TD0 & 0xFFFFu) << 16);              // atomic_bar=0 | tensor_dim0.lo16
        g1[2] = (int)(((TD0 >> 16) & 0xFFFFu)              // tensor_dim0.hi16
              | (1u << 16));                               // tensor_dim1 = 1 (lo16)
        g1[3] = (int)((TD0 & 0xFFFFu) << 16);              // tensor_dim1.hi16=0 | tile_dim0=8192
        g1[4] = 1;                                         // tile_dim1=1, tile_dim2=0
        g1[5] = (int)TD0;                                  // tensor_dim0_stride.lo32
        g1[6] = (int)((TD0 & 0xFFFFu) << 16);              // stride0.hi16=0 | stride1.lo16
        g1[7] = 0;                                         // stride1.hi32
        i32x4 gz4 = {0, 0, 0, 0};                          // groups 2/3: <=2-D tensor
        i32x8 gz8 = {0, 0, 0, 0, 0, 0, 0, 0};              // clang-23 6-arg form: extra group

        __builtin_amdgcn_tensor_load_to_lds(g0, g1, gz4, gz4, gz8, 0);
        __builtin_amdgcn_s_wait_tensorcnt(0);
    }
    __syncthreads();

    // ---- 1. build sort keys from LDS: key = (score > thr) ? score : -inf ---
    for (int p = tid; p < NPROP; p += NTHREAD) {
        float s  = s_scr[p * 2 + 1];
        s_key[p] = (s > SCORE_THRESH) ? s : -__builtin_inff();
        s_idx[p] = p;
    }
    __syncthreads();

    // ---- 2. bitonic sort: descending key, stable (ascending idx on ties) ----
    for (int k = 2; k <= NPROP; k <<= 1) {
        for (int j = k >> 1; j > 0; j >>= 1) {
            for (int t = tid; t < NPROP / 2; t += NTHREAD) {
                int i   = ((t & ~(j - 1)) << 1) | (t & (j - 1));
                int ixj = i | j;
                float ka = s_key[i], kb = s_key[ixj];
                int   ia = s_idx[i], ib = s_idx[ixj];
                bool before = (ka > kb) || (ka == kb && ia < ib); // a ranks first
                bool want   = ((i & k) == 0);
                if (before != want) {
                    s_key[i]   = kb; s_idx[i]   = ib;
                    s_key[ixj] = ka; s_idx[ixj] = ia;
                }
            }
            __syncthreads();
        }
    }

    // ---- 3. gather sorted boxes into LDS via async global->LDS (B128) -------
    for (int p = tid; p < NPROP; p += NTHREAD) {
        int o = s_idx[p];
        const float* g = props + (size_t)o * 4;
        unsigned lds_off = (unsigned)(uintptr_t)(&s_box[p][0]);
        asm volatile("global_load_async_to_lds_b128 %0, %1, off"
                     :: "v"(lds_off), "v"(g) : "memory");
    }
    asm volatile("s_wait_asynccnt 0x0" ::: "memory");
    __syncthreads();

    for (int p = tid; p < NPROP; p += NTHREAD) {
        s_area[p] = (s_box[p][2] - s_box[p][0]) * (s_box[p][3] - s_box[p][1]);
    }

    // ---- 4. init suppression bitmap: invalid (key <= thr) start suppressed --
    for (int p = tid; p < NPROP / 32; p += NTHREAD) s_supp[p] = 0u;
    __syncthreads();
    for (int p = tid; p < NPROP; p += NTHREAD) {
        if (!(s_key[p] > SCORE_THRESH))
            atomicOr(&s_supp[p >> 5], 1u << (p & 31));   // ds_or_b32
    }
    __syncthreads();

    // ---- 5. greedy NMS: serial over i, block-parallel over j > i ------------
    for (int i = 0; i < NPROP - 1; ++i) {
        bool keep_i = ((s_supp[i >> 5] >> (i & 31)) & 1u) == 0u;  // LDS broadcast
        __syncthreads();   // all reads of bit i done before this round's writes
        if (keep_i) {
            float xi1 = s_box[i][0], yi1 = s_box[i][1];
            float xi2 = s_box[i][2], yi2 = s_box[i][3];
            float ai  = s_area[i];
            for (int j = i + 1 + tid; j < NPROP; j += NTHREAD) {
                float xx1 = fmaxf(xi1, s_box[j][0]);
                float yy1 = fmaxf(yi1, s_box[j][1]);
                float xx2 = fminf(xi2, s_box[j][2]);
                float yy2 = fminf(yi2, s_box[j][3]);
                float w = fmaxf(xx2 - xx1, 0.0f);
                float h = fmaxf(yy2 - yy1, 0.0f);
                float inter = w * h;
                float iou = inter / (ai + s_area[j] - inter);
                if (iou >= IOU_THRESH)
                    atomicOr(&s_supp[j >> 5], 1u << (j & 31));
            }
        }
        __syncthreads();   // writes of round i visible before round i+1 reads
    }

    // ---- 6. compact kept positions, pad like the reference ------------------
    if (tid == 0) {
        int cnt = 0;
        for (int p = 0; p < NPROP; ++p) {
            if (((s_supp[p >> 5] >> (p & 31)) & 1u) == 0u) {
                if (cnt < MAXP) s_sel[cnt] = p;
                cnt++;
            }
        }
        if (cnt == 0) {
            for (int k2 = 0; k2 < MAXP; ++k2) s_sel[k2] = NPROP - 1; // clip(pos[0]=n)
        } else {
            int last = s_sel[(cnt < MAXP ? cnt : MAXP) - 1];
            for (int k2 = cnt; k2 < MAXP; ++k2) s_sel[k2] = last;
        }
    }
    __syncthreads();

    // ---- 7. write outputs: boxes [B,MAXP,4] then scores [B,MAXP] ------------
    float* out_boxes  = out;
    float* out_scores = out + (size_t)NBATCH * MAXP * 4;
    for (int k = tid; k < MAXP; k += NTHREAD) {
        int p = s_sel[k];
        int o = s_idx[p];
        size_t ob = ((size_t)b * MAXP + k) * 4;
        out_boxes[ob + 0] = s_box[p][0];
        out_boxes[ob + 1] = s_box[p][1];
        out_boxes[ob + 2] = s_box[p][2];
        out_boxes[ob + 3] = s_box[p][3];
        out_scores[(size_t)b * MAXP + k] = s_scr[o * 2 + 1];   // from TDM-staged LDS
    }
}

extern "C" void kernel_launch(void* const* d_in, const int* in_sizes, int n_in,
                              void* d_out, int out_size, void* d_ws, size_t ws_size,
                              hipStream_t stream) {
    (void)in_sizes; (void)n_in; (void)d_ws; (void)ws_size; (void)out_size;
    const float* proposals  = (const float*)d_in[0];
    const float* cls_scores = (const float*)d_in[1];
    nms_gfx1250_kernel<<<dim3(NBATCH), dim3(NTHREAD), 0, stream>>>(
        proposals, cls_scores, (float*)d_out);
}